// GaussianSplat2D_81621558493558
// MI455X (gfx1250) — compile-verified
//
#include <hip/hip_runtime.h>
#include <math.h>

// CDNA5 / gfx1250: wave32, WMMA f32 16x16x4.
typedef __attribute__((ext_vector_type(2))) float v2f;
typedef __attribute__((ext_vector_type(8))) float v8f;

#define NG 512
#define HH 256
#define WW 256

// -0.5 * log2(e): folded into the gaussian coefficients so that the WMMA
// output is directly the base-2 exponent (v_exp_f32 computes 2^x natively).
#define QSCALE (-0.72134752044448170368f)

// ---------------------------------------------------------------------------
// Prep: per-gaussian polynomial coefficients for the expanded quadratic form
//   quad_n(x,y) = ia x^2 + 2ib xy + ic y^2 - 2(ia mx + ib my) x
//                 - 2(ib mx + ic my) y + (ia mx^2 + 2ib mx my + ic my^2)
// (coordinates centered at 128 for fp32 conditioning). All coefficients are
// pre-scaled by -0.5*log2(e) so exp(-0.5*quad) == exp2(WMMA output).
// Also: c4_n = amp * [1, r, g, b],  amp = opacity / (2*pi*sqrt(det + 1e-8))
// ---------------------------------------------------------------------------
__global__ void splat_prep(const float* __restrict__ means,
                           const float* __restrict__ covs,
                           const float* __restrict__ colors,
                           const float* __restrict__ opac,
                           float* __restrict__ gcoef,   // [NG][8]
                           float* __restrict__ c4)      // [NG][4]
{
    int n = blockIdx.x * blockDim.x + threadIdx.x;
    if (n >= NG) return;
    float mx = means[2*n+0] - 128.0f;   // x (column)
    float my = means[2*n+1] - 128.0f;   // y (row)
    float a = covs[4*n+0], b = covs[4*n+1], c = covs[4*n+3];
    float det = a*c - b*b;
    float inv_det = 1.0f / det;
    float ia =  c*inv_det;
    float ib = -b*inv_det;
    float ic =  a*inv_det;
    float amp = opac[n] / (6.28318530717958647f * sqrtf(det + 1e-8f));

    float* g = gcoef + 8*n;
    g[0] = QSCALE * ia;
    g[1] = QSCALE * 2.0f*ib;
    g[2] = QSCALE * ic;
    g[3] = QSCALE * -2.0f*(ia*mx + ib*my);
    g[4] = QSCALE * -2.0f*(ib*mx + ic*my);
    g[5] = QSCALE * fmaf(ia*mx, mx, fmaf(2.0f*ib*mx, my, ic*my*my));
    g[6] = 0.0f;
    g[7] = 0.0f;

    float* cc = c4 + 4*n;
    cc[0] = amp;
    cc[1] = amp * colors[3*n+0];
    cc[2] = amp * colors[3*n+1];
    cc[3] = amp * colors[3*n+2];
}

// ---------------------------------------------------------------------------
// Main fused splat kernel.
// Block = 256 threads = 8 waves. Wave w handles a strip of 16 pixels in one
// image row. Per 16-gaussian tile: scaled quad(16px x 16g) via two chained
// V_WMMA_F32_16X16X4_F32 (K = 8 features), then v_exp_f32 directly (scale
// pre-folded), and per-lane dual-FMA accumulation of [sum_w, r, g, b];
// final 16-lane xor-shuffle reduction over gaussian columns.
// ---------------------------------------------------------------------------
__global__ __launch_bounds__(256)
void splat_main(const float* __restrict__ gcoef_g,
                const float4* __restrict__ c4_g,
                float* __restrict__ out)
{
    __shared__ float  sG[NG * 8];   // 16 KB
    __shared__ float4 sC[NG];       //  8 KB

    const int tid = threadIdx.x;
    for (int i = tid; i < NG * 8; i += 256) sG[i] = gcoef_g[i];
    for (int i = tid; i < NG;     i += 256) sC[i] = c4_g[i];
    __syncthreads();

    const int lane = tid & 31;
    const int wave = tid >> 5;
    const int strip = blockIdx.x * 8 + wave;     // 4096 strips of 16 pixels
    const int row   = strip >> 4;                // y
    const int x0    = (strip & 15) << 4;         // x base

    const int half = lane >> 4;                  // K-half (A layout) / M-half (D layout)
    const int m    = lane & 15;                  // pixel row M (A) / gaussian col N (B,D)

    const float fx = (float)(x0 + m) - 128.0f;
    const float fy = (float)row - 128.0f;

    // A fragments (16x4 f32 layout: lane m holds K = 2*half + {0,1}).
    // features K0..K7 = [x^2, x*y, y^2, x, y, 1, 0, 0]
    v2f a1, a2;
    a1.x = half ? (fy * fy) : (fx * fx);
    a1.y = half ? fx        : (fx * fy);
    a2.x = half ? 0.0f      : fy;
    a2.y = half ? 0.0f      : 1.0f;

    float4 acc[8];
#pragma unroll
    for (int r = 0; r < 8; ++r) acc[r] = make_float4(0.f, 0.f, 0.f, 0.f);

    const int k0 = half << 1;

    for (int t = 0; t < NG / 16; ++t) {
        const int n = (t << 4) + m;              // this lane's gaussian column
        const float* g = &sG[n * 8];
        // B fragments (4x16 f32, mirrored A layout): lane n holds K = 2*half+{0,1}
        v2f b1 = *(const v2f*)&g[k0];            // features k0, k0+1
        v2f b2 = *(const v2f*)&g[4 + k0];        // features 4+k0, 5+k0

        v8f q = {};
        q = __builtin_amdgcn_wmma_f32_16x16x4_f32(false, a1, false, b1,
                                                  (short)0, q, false, false);
        q = __builtin_amdgcn_wmma_f32_16x16x4_f32(false, a2, false, b2,
                                                  (short)0, q, false, false);

        const float4 cn = sC[n];                 // amp * [1, r, g, b]
#pragma unroll
        for (int r = 0; r < 8; ++r) {            // pixel M = r + 8*half
            float e = __builtin_amdgcn_exp2f(q[r]);   // v_exp_f32, no pre-scale
            acc[r].x = fmaf(e, cn.x, acc[r].x);
            acc[r].y = fmaf(e, cn.y, acc[r].y);
            acc[r].z = fmaf(e, cn.z, acc[r].z);
            acc[r].w = fmaf(e, cn.w, acc[r].w);
        }
    }

    // Reduce over the 16 gaussian columns held by each 16-lane half
    // (xor masks 1,2,4,8 stay inside each half).
#pragma unroll
    for (int mask = 1; mask <= 8; mask <<= 1) {
#pragma unroll
        for (int r = 0; r < 8; ++r) {
            acc[r].x += __shfl_xor(acc[r].x, mask, 32);
            acc[r].y += __shfl_xor(acc[r].y, mask, 32);
            acc[r].z += __shfl_xor(acc[r].z, mask, 32);
            acc[r].w += __shfl_xor(acc[r].w, mask, 32);
        }
    }

    // Lanes m<8 of each half write one pixel each (select keeps acc in VGPRs).
    if (m < 8) {
        float4 v = acc[0];
#pragma unroll
        for (int r = 1; r < 8; ++r) if (m == r) v = acc[r];
        const int x = x0 + (half << 3) + m;      // pixel M = m + 8*half
        const float inv = 1.0f / fmaxf(v.x, 1e-8f);
        float* o = out + (row * WW + x) * 3;
        o[0] = v.y * inv;
        o[1] = v.z * inv;
        o[2] = v.w * inv;
    }
}

extern "C" void kernel_launch(void* const* d_in, const int* in_sizes, int n_in,
                              void* d_out, int out_size, void* d_ws, size_t ws_size,
                              hipStream_t stream) {
    const float* means  = (const float*)d_in[0];   // (N,2) f32
    const float* covs   = (const float*)d_in[1];   // (N,2,2) f32
    const float* colors = (const float*)d_in[2];   // (N,3) f32
    const float* opac   = (const float*)d_in[3];   // (N,) f32
    // d_in[4]=height, d_in[5]=width (compile-time constants here)

    float* gcoef = (float*)d_ws;                   // NG*8 floats = 16 KB
    float* c4    = (float*)d_ws + NG * 8;          // NG*4 floats =  8 KB (16B aligned)

    splat_prep<<<(NG + 255) / 256, 256, 0, stream>>>(means, covs, colors, opac,
                                                     gcoef, c4);
    splat_main<<<(HH * WW) / (8 * 16), 256, 0, stream>>>(gcoef,
                                                         (const float4*)c4,
                                                         (float*)d_out);
}